// SA_global_shifted_Swin_40415642255817
// MI455X (gfx1250) — compile-verified
//
#include <hip/hip_runtime.h>

// ---------------------------------------------------------------------------
// Fused shifted-window attention for MI455X (gfx1250, wave32).
// One workgroup (256 thr = 8 waves) handles one (batch, window, half).
//   0. Wave 0 issues a Tensor-Data-Mover DMA (tensor_load_to_lds) staging the
//      192x64 f32 qkv_w into LDS with a 1-DWORD pad every 64 DWORDs
//      (row stride 65 -> bank-conflict-free WMMA A-operand gathers).
//   1. Stage x 6x6 halo (64ch, rolled, zero-padded), bias table, lepe weights.
//   2. WMMA f32 16x16x4 projection: q,k for the 16 window pixels, v for the
//      full 6x6 halo (zero-padded borders -> zero v, matches SAME conv pad).
//   3. Per-head (1 head per wave): S = scale*Q K^T + bias (2 WMMA),
//      row softmax via 16-lane shfl_xor butterflies, O = P V (4 WMMA).
//   4. LePE 3x3 depthwise from v-halo, add, store with inverse roll.
// Launch twice: shift=0 (store) then shift=2 (read-modify-write add).
// ---------------------------------------------------------------------------

typedef float v2f __attribute__((ext_vector_type(2)));
typedef float v8f __attribute__((ext_vector_type(8)));
typedef unsigned int v4u __attribute__((ext_vector_type(4)));
typedef int v8i __attribute__((ext_vector_type(8)));
typedef int v4i __attribute__((ext_vector_type(4)));

#define SPD   4
#define HEADS 8
#define CHALF 64
#define HW    192
#define NWIN  48
#define BATCH 8

__device__ __forceinline__ v8f wmma4(v2f a, v2f b, v8f c) {
  // D = A(16x4) * B(4x16) + C   (fp32 WMMA)
  return __builtin_amdgcn_wmma_f32_16x16x4_f32(false, a, false, b, (short)0, c,
                                               false, false);
}

// window pixel p (0..15) -> index into 6x6 halo (row-major, +1,+1 offset)
__device__ __forceinline__ int hidx(int p) {
  return ((p >> 2) + 1) * 6 + (p & 3) + 1;
}

__global__ __launch_bounds__(256)
void swin_fused_attn(const float* __restrict__ x,
                     const float* __restrict__ qkv_w,
                     const float* __restrict__ lepe_w,
                     const float* __restrict__ bias_table,
                     float* __restrict__ out,
                     int shift, int accum)
{
  __shared__ float sW[192 * 65];      // qkv_w, row-padded (49.9 KB), TDM-filled
  __shared__ float sXh[CHALF * 37];   // x halo [ci][6x6], padded
  __shared__ float sQ[CHALF * 17];    // q [ch][pos]
  __shared__ float sK[CHALF * 17];    // k [ch][pos]
  __shared__ float sVh[CHALF * 37];   // v halo [ch][6x6]
  __shared__ float sP[8 * 16 * 17];   // per-wave softmaxed attn 16x16
  __shared__ float sBias[49 * 8];
  __shared__ float sLepeW[64 * 9];

  const int tid   = threadIdx.x;
  const int wv    = tid >> 5;     // wave id 0..7
  const int lane  = tid & 31;
  const int l15   = lane & 15;
  const int khalf = lane >> 4;    // 0: lanes 0-15, 1: lanes 16-31

  int t = blockIdx.x;
  const int half = t & 1; t >>= 1;
  const int win  = t % (NWIN * NWIN);
  const int b    = t / (NWIN * NWIN);
  const int wy   = win / NWIN, wx = win % NWIN;

  // ---- TDM: DMA qkv_w (192x64 f32) -> sW with row stride 65 ---------------
  // D# group0: count=1 | lds_addr | global_addr(57b) | type=2
  // D# group1: data_size=4B, pad_enable, pad_interval=5 (64 DWORDs),
  //            pad_amount=0 (1 DWORD), 1-D tile of 12288 elements.
  if (wv == 0) {
    unsigned long long ga = (unsigned long long)(uintptr_t)qkv_w;
    unsigned int ldsoff = (unsigned int)(uintptr_t)(&sW[0]);  // low 32 = LDS offset
    v4u g0;
    g0.x = 1u;                                   // count=1, user descriptor
    g0.y = ldsoff;                               // lds_addr
    g0.z = (unsigned int)(ga & 0xffffffffu);     // global_addr[31:0]
    g0.w = (unsigned int)((ga >> 32) & 0x1ffffffu) | 0x80000000u; // [56:32]|type=2
    v8i g1;
    g1[0] = (2 << 16)        // data_size = 4 bytes
          | (1 << 20)        // pad_enable
          | (5 << 22);       // pad_interval: 64 DWORDs ; pad_amount=0 -> 1 DWORD
    g1[1] = (12288 << 16);   // tensor_dim0[15:0]  (=0x3000)
    g1[2] = (1 << 16);       // tensor_dim0[31:16]=0 | tensor_dim1[15:0]=1
    g1[3] = (12288 << 16);   // tensor_dim1[31:16]=0 | tile_dim0 = 12288
    g1[4] = 1;               // tile_dim1 = 1, tile_dim2 = 0
    g1[5] = 12288;           // tensor_dim0_stride[31:0]
    g1[6] = (12288 << 16);   // stride[47:32]=0 | tensor_dim1_stride[15:0]
    g1[7] = 0;               // tensor_dim1_stride[47:16]
    v4i gz4 = {0, 0, 0, 0};            // groups 2/3 unused (<=2-D tensor)
    v8i gz8 = {0, 0, 0, 0, 0, 0, 0, 0};
    __builtin_amdgcn_tensor_load_to_lds(g0, g1, gz4, gz4, gz8, 0);
    __builtin_amdgcn_s_wait_tensorcnt(0);
  }

  // ---- stage small tables -------------------------------------------------
  if (tid < 49 * 8) sBias[tid]  = bias_table[tid];
  if (tid < 64 * 9) sLepeW[tid] = lepe_w[tid];

  // ---- stage rolled x halo: 64 ch x 6x6, zero outside rolled-frame bounds --
#pragma unroll
  for (int it = 0; it < 9; ++it) {
    int idx = tid + it * 256;                  // 0..2303
    int ci = idx / 36, hp = idx % 36;
    int hy = hp / 6, hx2 = hp % 6;
    int hr = wy * SPD - 1 + hy;                // rolled-frame row
    int wc = wx * SPD - 1 + hx2;               // col (W never rolled)
    float v = 0.f;
    if (hr >= 0 && hr < HW && wc >= 0 && wc < HW) {
      int cg = (half * CHALF + ci + shift) & 127;  // roll on channel
      int rg = (hr + shift) % HW;                  // roll on H
      v = x[(((size_t)b * 128 + cg) * HW + rg) * HW + wc];
    }
    sXh[ci * 37 + hp] = v;
  }
  __syncthreads();

  // ---- projection: q,k for window pixels (tile m = wave) -------------------
  {
    const int obase = wv * 16;                 // o rows of this wave's tile
    v8f acc = {};
#pragma unroll
    for (int kk = 0; kk < 16; ++kk) {
      int kb = kk * 4 + 2 * khalf;
      v2f a, bb;
      a.x  = sW[(obase + l15) * 65 + kb];
      a.y  = sW[(obase + l15) * 65 + kb + 1];
      bb.x = sXh[kb * 37 + hidx(l15)];
      bb.y = sXh[(kb + 1) * 37 + hidx(l15)];
      acc = wmma4(a, bb, acc);
    }
    float* dst = (wv < 4) ? sQ : sK;
    const int ob = obase & 63;
#pragma unroll
    for (int r = 0; r < 8; ++r)
      dst[(ob + r + 8 * khalf) * 17 + l15] = acc[r];
  }
  // ---- projection: v over the 36-pixel halo (12 tiles over 8 waves) --------
  for (int tt = wv; tt < 12; tt += 8) {
    const int vm = tt / 3, nt = tt % 3;
    const int obase = 128 + vm * 16;           // v rows in qkv_w
    const int cb = nt * 16;
    v8f acc = {};
#pragma unroll
    for (int kk = 0; kk < 16; ++kk) {
      int kb = kk * 4 + 2 * khalf;
      int col = cb + l15;
      v2f a, bb;
      a.x  = sW[(obase + l15) * 65 + kb];
      a.y  = sW[(obase + l15) * 65 + kb + 1];
      bb.x = (col < 36) ? sXh[kb * 37 + col] : 0.f;
      bb.y = (col < 36) ? sXh[(kb + 1) * 37 + col] : 0.f;
      acc = wmma4(a, bb, acc);
    }
    const int col = cb + l15;
#pragma unroll
    for (int r = 0; r < 8; ++r)
      if (col < 36) sVh[(vm * 16 + r + 8 * khalf) * 37 + col] = acc[r];
  }
  __syncthreads();

  // ---- attention: one head per wave ---------------------------------------
  const float scale = 0.35355339059327373f;    // 1/sqrt(hd=8)
  const int hh = wv;                           // head; channel = f*8 + hh
  v8f sacc = {};
#pragma unroll
  for (int kk = 0; kk < 2; ++kk) {             // K = hd = 8 -> 2 steps
    int f0 = kk * 4 + 2 * khalf;
    v2f a, bb;
    a.x  = sQ[(f0 * 8 + hh) * 17 + l15] * scale;
    a.y  = sQ[((f0 + 1) * 8 + hh) * 17 + l15] * scale;
    bb.x = sK[(f0 * 8 + hh) * 17 + l15];
    bb.y = sK[((f0 + 1) * 8 + hh) * 17 + l15];
    sacc = wmma4(a, bb, sacc);
  }
  // bias + row softmax (row = 16 lanes of one half-wave)
  float p[8];
  {
    const int kr = l15 >> 2, kc = l15 & 3;     // key pixel coords
#pragma unroll
    for (int r = 0; r < 8; ++r) {
      int qpos = r + 8 * khalf;
      int qr = qpos >> 2, qc = qpos & 3;
      int idx = (qr - kr + 3) * 7 + (qc - kc + 3);
      p[r] = sacc[r] + sBias[idx * 8 + hh];
    }
#pragma unroll
    for (int r = 0; r < 8; ++r) {
      float m = p[r];
#pragma unroll
      for (int msk = 8; msk >= 1; msk >>= 1)
        m = fmaxf(m, __shfl_xor(m, msk, 32));
      float e = __expf(p[r] - m);
      float ssum = e;
#pragma unroll
      for (int msk = 8; msk >= 1; msk >>= 1)
        ssum += __shfl_xor(ssum, msk, 32);
      p[r] = e / ssum;
    }
  }
  float* myP = &sP[wv * 16 * 17];
#pragma unroll
  for (int r = 0; r < 8; ++r)
    myP[(r + 8 * khalf) * 17 + l15] = p[r];

  // O = P @ V  (K = 16 key positions -> 4 steps; N cols 8..15 are padding)
  v8f oacc = {};
  {
    const int chn = (l15 & 7) * 8 + hh;        // safe even for pad lanes
    const bool valn = (l15 < 8);
#pragma unroll
    for (int kk = 0; kk < 4; ++kk) {
      int kb = kk * 4 + 2 * khalf;
      v2f a, bb;
      a.x  = myP[l15 * 17 + kb];
      a.y  = myP[l15 * 17 + kb + 1];
      bb.x = valn ? sVh[chn * 37 + hidx(kb)]     : 0.f;
      bb.y = valn ? sVh[chn * 37 + hidx(kb + 1)] : 0.f;
      oacc = wmma4(a, bb, oacc);
    }
  }

  // ---- LePE + store with inverse roll -------------------------------------
  if (l15 < 8) {
    const int ch = l15 * 8 + hh;               // channel within half
    const float* wp = &sLepeW[ch * 9];
#pragma unroll
    for (int r = 0; r < 8; ++r) {
      int qpos = r + 8 * khalf;
      int ir = qpos >> 2, ic = qpos & 3;
      float lepe = 0.f;
#pragma unroll
      for (int dy = 0; dy < 3; ++dy)
#pragma unroll
        for (int dx = 0; dx < 3; ++dx)
          lepe += wp[dy * 3 + dx] * sVh[ch * 37 + (ir + dy) * 6 + (ic + dx)];
      float val = oacc[r] + lepe;
      int hr = wy * SPD + ir;
      int wc = wx * SPD + ic;
      int orow  = (hr + shift) % HW;                    // un-roll H
      int ochan = (half * CHALF + ch + shift) & 127;    // un-roll channel
      size_t oidx = (((size_t)b * 128 + ochan) * HW + orow) * HW + wc;
      if (accum) out[oidx] += val;
      else       out[oidx]  = val;
    }
  }
}

extern "C" void kernel_launch(void* const* d_in, const int* in_sizes, int n_in,
                              void* d_out, int out_size, void* d_ws, size_t ws_size,
                              hipStream_t stream) {
  (void)in_sizes; (void)n_in; (void)out_size; (void)d_ws; (void)ws_size;
  const float* x          = (const float*)d_in[0];
  const float* qkv_w      = (const float*)d_in[1];
  const float* lepe_w     = (const float*)d_in[2];
  const float* bias_table = (const float*)d_in[3];
  float* out = (float*)d_out;

  dim3 grid(BATCH * NWIN * NWIN * 2);
  dim3 blk(256);
  // shift 0: full overwrite of d_out; shift 2: accumulate (stream-ordered)
  hipLaunchKernelGGL(swin_fused_attn, grid, blk, 0, stream,
                     x, qkv_w, lepe_w, bias_table, out, 0, 0);
  hipLaunchKernelGGL(swin_fused_attn, grid, blk, 0, stream,
                     x, qkv_w, lepe_w, bias_table, out, 2, 1);
}